// ToyTransformerBlock_4999341932774
// MI455X (gfx1250) — compile-verified
//
#include <hip/hip_runtime.h>
#include <hip/hip_bf16.h>
#include <math.h>

// ---------------------------------------------------------------------------
// ToyTransformerBlock on MI455X (gfx1250, wave32).
// Roofline: ~137 GFLOP vs ~80MB HBM traffic -> memory bound at 23.3 TB/s.
// => keep full f32 precision and use V_WMMA_F32_16X16X4_F32 (matrix pipe,
//    2-VGPR A/B fragments, f32 accumulation) for every GEMM stage.
// ---------------------------------------------------------------------------

typedef __attribute__((ext_vector_type(2))) float v2f;
typedef __attribute__((ext_vector_type(8))) float v8f;

#define BB 4
#define SS 2048
#define DD 1024

// D = A(16x4) * B(4x16) + C, f32.  8-arg VOP3P builtin form.
__device__ __forceinline__ v8f wmma4(v2f a, v2f b, v8f c) {
  return __builtin_amdgcn_wmma_f32_16x16x4_f32(
      /*neg_a=*/false, a, /*neg_b=*/false, b,
      /*c_mod=*/(short)0, c, /*reuse_a=*/false, /*reuse_b=*/false);
}

// A-fragment (16x4, rows m0..m0+15, cols k0..k0+3) from row-major [.,ld]:
// lane (lo,hi): { M[(m0+lo)][k0+2hi], M[(m0+lo)][k0+2hi+1] }  (8B aligned)
// Same pattern serves the B-fragment of y = x*W^T (B[k][n] = W[n][k]):
// lane reads row n0+lo of W at col k0+2hi. (CDNA5 ISA 7.12.2)
__device__ __forceinline__ v2f frag_rowk(const float* __restrict__ p,
                                         int row0, int k0, int ld,
                                         int lo, int hi) {
  const float* q = p + (size_t)(row0 + lo) * ld + (k0 + 2 * hi);
  return v2f{q[0], q[1]};
}

// B-fragment (4x16) straight from a row-major matrix: B[k][n] = M[k0+k][n0+n]
// lane (lo,hi): { M[k0+2hi][n0+lo], M[k0+2hi+1][n0+lo] }  (stride-ld pair)
__device__ __forceinline__ v2f fragB_direct(const float* __restrict__ p,
                                            int k0, int n0, int ld,
                                            int lo, int hi) {
  const float* q = p + (size_t)(k0 + 2 * hi) * ld + (n0 + lo);
  return v2f{q[0], q[(size_t)ld]};
}

// ---------------------------------------------------------------------------
// Kernel 1: Q/K/V projection.  grid = (BS/16, 3), block = 256 (8 waves).
// Each wave: 16(M) x 128(N) tile, 8 v8f accumulators, K-loop over D step 4.
// ---------------------------------------------------------------------------
__global__ __launch_bounds__(256) void k_proj(
    const float* __restrict__ x,
    const float* __restrict__ Wq, const float* __restrict__ Wk,
    const float* __restrict__ Wv,
    float* __restrict__ Q, float* __restrict__ K, float* __restrict__ V) {
  const int w = threadIdx.x >> 5;
  const int lane = threadIdx.x & 31, lo = lane & 15, hi = lane >> 4;
  const int m0 = blockIdx.x * 16;
  const int n0 = w * 128;

  const float* W;
  float* out;
  if (blockIdx.y == 0)      { W = Wq; out = Q; }
  else if (blockIdx.y == 1) { W = Wk; out = K; }
  else                      { W = Wv; out = V; }

  v8f acc[8] = {};
  for (int k0 = 0; k0 < DD; k0 += 4) {
    v2f a = frag_rowk(x, m0, k0, DD, lo, hi);      // x tile (shared across n)
#pragma unroll
    for (int i = 0; i < 8; ++i) {
      v2f b = frag_rowk(W, n0 + i * 16, k0, DD, lo, hi);  // W rows (y=xW^T)
      acc[i] = wmma4(a, b, acc[i]);
    }
  }
#pragma unroll
  for (int i = 0; i < 8; ++i) {
#pragma unroll
    for (int r = 0; r < 8; ++r) {
      out[(size_t)(m0 + r + 8 * hi) * DD + n0 + i * 16 + lo] = acc[i][r];
    }
  }
}

// ---------------------------------------------------------------------------
// Kernel 2: flash attention (online softmax).  grid = B*(S/16), block = 256.
// Block owns one (batch, 16-query) tile.  Wave w owns D-slice [128w,128w+128):
//  - 32 pre-loaded Q A-fragments (64 VGPRs)
//  - partial score tile via 32 WMMAs; cross-wave sum in LDS
//  - softmax with running (m, l); rescale accumulators by alpha
//  - probs @ V: 4 K-steps x 8 N-subtiles WMMAs into 8 v8f accumulators
// ---------------------------------------------------------------------------
__global__ __launch_bounds__(256) void k_attn(
    const float* __restrict__ Q, const float* __restrict__ K,
    const float* __restrict__ V, float* __restrict__ ATT) {
  __shared__ float scpart[8 * 256];   // per-wave partial score tiles
  __shared__ float sfull[256];        // summed scores (16x16)
  __shared__ float pbuf[256];         // exp probs (16x16)
  __shared__ float m_run[16], l_run[16], albuf[16];

  const int t = threadIdx.x;
  const int w = t >> 5, lane = t & 31, lo = lane & 15, hi = lane >> 4;
  const int b = blockIdx.x >> 7;          // batch
  const int qt = blockIdx.x & 127;        // query tile
  const int m0 = qt * 16;
  const int d0 = w * 128;

  const float* Qb = Q + (size_t)b * SS * DD;
  const float* Kb = K + (size_t)b * SS * DD;
  const float* Vb = V + (size_t)b * SS * DD;

  // Pre-load Q fragments for this wave's D-slice.
  v2f qa[32];
#pragma unroll
  for (int kk = 0; kk < 32; ++kk)
    qa[kk] = frag_rowk(Qb, m0, d0 + kk * 4, DD, lo, hi);

  v8f acc[8] = {};
  if (t < 16) { m_run[t] = -1e30f; l_run[t] = 0.0f; }
  __syncthreads();

  const float scale = 0.03125f;  // 1/sqrt(1024)

  for (int j = 0; j < SS / 16; ++j) {
    const float* Kt = Kb + (size_t)(j * 16) * DD;

    // --- partial scores over this wave's D-slice ---------------------------
    v8f s = {};
#pragma unroll
    for (int kk = 0; kk < 32; ++kk) {
      // scores[m][n] = sum_d q[m,d]*k[n,d]  => B[k][n] = K[row n][d] (row-k)
      v2f kb = frag_rowk(Kt, 0, d0 + kk * 4, DD, lo, hi);
      s = wmma4(qa[kk], kb, s);
    }
#pragma unroll
    for (int r = 0; r < 8; ++r)
      scpart[w * 256 + (r + 8 * hi) * 16 + lo] = s[r];
    __syncthreads();

    // --- cross-wave sum + online softmax (256 threads <-> 16x16 tile) ------
    const int m = t >> 4, n = t & 15;
    float sv = 0.f;
#pragma unroll
    for (int ww = 0; ww < 8; ++ww) sv += scpart[ww * 256 + t];
    sv *= scale;
    sfull[t] = sv;
    __syncthreads();

    float m_old = m_run[m];
    float m_new = m_old;
#pragma unroll
    for (int c = 0; c < 16; ++c) m_new = fmaxf(m_new, sfull[m * 16 + c]);
    float p = __expf(sv - m_new);
    pbuf[t] = p;
    if (n == 0) albuf[m] = __expf(m_old - m_new);
    __syncthreads();

    float rowsum = 0.f;
#pragma unroll
    for (int c = 0; c < 16; ++c) rowsum += pbuf[m * 16 + c];
    if (n == 0) {
      l_run[m] = l_run[m] * albuf[m] + rowsum;
      m_run[m] = m_new;
    }

    // --- rescale running accumulators by alpha per row ---------------------
#pragma unroll
    for (int r = 0; r < 8; ++r) {
      float al = albuf[r + 8 * hi];
#pragma unroll
      for (int i = 0; i < 8; ++i) acc[i][r] *= al;
    }

    // --- attended += probs @ V (A from LDS prob tile, B direct from V) -----
    const float* Vt = Vb + (size_t)(j * 16) * DD;
#pragma unroll
    for (int kk = 0; kk < 4; ++kk) {
      const float* pp = &pbuf[lo * 16 + kk * 4 + 2 * hi];
      v2f pa = v2f{pp[0], pp[1]};
#pragma unroll
      for (int i = 0; i < 8; ++i) {
        v2f vb = fragB_direct(Vt, kk * 4, d0 + i * 16, DD, lo, hi);
        acc[i] = wmma4(pa, vb, acc[i]);
      }
    }
    __syncthreads();  // protect pbuf/albuf/l_run before next tile overwrites
  }

  // --- final 1/l normalization and store ----------------------------------
  float* Ab = ATT + (size_t)b * SS * DD;
#pragma unroll
  for (int r = 0; r < 8; ++r) {
    float inv = 1.0f / l_run[r + 8 * hi];
    const int row = m0 + r + 8 * hi;
#pragma unroll
    for (int i = 0; i < 8; ++i)
      Ab[(size_t)row * DD + d0 + i * 16 + lo] = acc[i][r] * inv;
  }
}

// ---------------------------------------------------------------------------
// Kernel 3: output projection + residual + LayerNorm.  grid = BS/16, blk 256.
// Same WMMA structure as k_proj; then residual in registers, LN via shfl_xor
// (within 16-lane half) + tiny cross-wave LDS reduction.
// ---------------------------------------------------------------------------
__global__ __launch_bounds__(256) void k_outproj_ln(
    const float* __restrict__ ATT, const float* __restrict__ Wo,
    const float* __restrict__ x, float* __restrict__ out) {
  __shared__ float psum[8 * 16], psq[8 * 16], mu[16], rs[16];

  const int w = threadIdx.x >> 5;
  const int lane = threadIdx.x & 31, lo = lane & 15, hi = lane >> 4;
  const int m0 = blockIdx.x * 16;
  const int n0 = w * 128;

  v8f acc[8] = {};
  for (int k0 = 0; k0 < DD; k0 += 4) {
    v2f a = frag_rowk(ATT, m0, k0, DD, lo, hi);
#pragma unroll
    for (int i = 0; i < 8; ++i) {
      v2f b = frag_rowk(Wo, n0 + i * 16, k0, DD, lo, hi);  // mixed = att*Wo^T
      acc[i] = wmma4(a, b, acc[i]);
    }
  }

  // residual add + per-row partial mean / meansq
#pragma unroll
  for (int r = 0; r < 8; ++r) {
    const int row = m0 + r + 8 * hi;
    float s = 0.f, sq = 0.f;
#pragma unroll
    for (int i = 0; i < 8; ++i) {
      float h = acc[i][r] + x[(size_t)row * DD + n0 + i * 16 + lo];
      acc[i][r] = h;
      s += h;
      sq += h * h;
    }
    // reduce over the 16 lanes of this half (bit4 of lane untouched)
#pragma unroll
    for (int off = 1; off < 16; off <<= 1) {
      s += __shfl_xor(s, off, 32);
      sq += __shfl_xor(sq, off, 32);
    }
    if (lo == 0) {
      psum[w * 16 + r + 8 * hi] = s;
      psq[w * 16 + r + 8 * hi] = sq;
    }
  }
  __syncthreads();

  if (threadIdx.x < 16) {
    float s = 0.f, sq = 0.f;
#pragma unroll
    for (int ww = 0; ww < 8; ++ww) {
      s += psum[ww * 16 + threadIdx.x];
      sq += psq[ww * 16 + threadIdx.x];
    }
    float mean = s * (1.0f / DD);
    float var = sq * (1.0f / DD) - mean * mean;
    mu[threadIdx.x] = mean;
    rs[threadIdx.x] = rsqrtf(var + 1e-5f);
  }
  __syncthreads();

#pragma unroll
  for (int r = 0; r < 8; ++r) {
    const int row = m0 + r + 8 * hi;
    const float mm = mu[r + 8 * hi], rr = rs[r + 8 * hi];
#pragma unroll
    for (int i = 0; i < 8; ++i)
      out[(size_t)row * DD + n0 + i * 16 + lo] = (acc[i][r] - mm) * rr;
  }
}

// ---------------------------------------------------------------------------
// Launch: Q, K, V, ATT scratch = 4 * B*S*D * 4B = 128 MB in d_ws.
// ---------------------------------------------------------------------------
extern "C" void kernel_launch(void* const* d_in, const int* in_sizes, int n_in,
                              void* d_out, int out_size, void* d_ws,
                              size_t ws_size, hipStream_t stream) {
  const float* x  = (const float*)d_in[0];
  const float* Wq = (const float*)d_in[1];
  const float* Wk = (const float*)d_in[2];
  const float* Wv = (const float*)d_in[3];
  const float* Wo = (const float*)d_in[4];
  float* out = (float*)d_out;

  float* ws = (float*)d_ws;
  const size_t NTOK = (size_t)BB * SS * DD;  // 8M elements
  float* Q   = ws + 0 * NTOK;
  float* K   = ws + 1 * NTOK;
  float* V   = ws + 2 * NTOK;
  float* ATT = ws + 3 * NTOK;

  const int MT = (BB * SS) / 16;  // 512 row tiles
  dim3 blk(256);
  k_proj<<<dim3(MT, 3), blk, 0, stream>>>(x, Wq, Wk, Wv, Q, K, V);
  k_attn<<<dim3(BB * (SS / 16)), blk, 0, stream>>>(Q, K, V, ATT);
  k_outproj_ln<<<dim3(MT), blk, 0, stream>>>(ATT, Wo, x, out);
}